// InvWaveletTransformLayer_82849919140040
// MI455X (gfx1250) — compile-verified
//
#include <hip/hip_runtime.h>
#include <stdint.h>

// Identity copy: out[i] = x[i], 64 * 2^20 float32.
// Pure HBM-bandwidth-bound (537 MB total traffic -> ~23 us floor at 23.3 TB/s).
// Implemented with the CDNA5 async global<->LDS streaming path:
//   GLOBAL_LOAD_ASYNC_TO_LDS_B128  (ASYNCcnt)
//   s_wait_asynccnt 0
//   GLOBAL_STORE_ASYNC_FROM_LDS_B128 (ASYNCcnt)
//   s_wait_asynccnt 0
// Each wave moves 8 KB per phase (16 x 512 B); latency is hidden by ~32K
// concurrent wave32s (4096 blocks x 8 waves; 40 waves resident per WGP at
// 64 KB LDS / block out of 320 KB).

#define THREADS 256
#define KBATCH 16            // float4 per thread per tile
#define TILE4 (THREADS * KBATCH)  // float4 elements per block-tile (4096 -> 64 KB)

__global__ __launch_bounds__(THREADS) void copy_async_lds_kernel(
    const float4* __restrict__ src, float4* __restrict__ dst, long long n4) {
  __shared__ float4 buf[TILE4];  // 64 KB staging buffer

  const unsigned t = threadIdx.x;
  const long long tileBase = (long long)blockIdx.x * TILE4;

  if (tileBase + TILE4 <= n4) {
    // ---- fast path: full tile, EXEC all-ones, async DMA through LDS ----
#pragma unroll
    for (int k = 0; k < KBATCH; ++k) {
      const unsigned idx = k * THREADS + t;
      // Low 32 bits of the generic pointer are the LDS byte address
      // (ISA 10.2, LDS aperture).
      const unsigned ldsAddr = (unsigned)(uintptr_t)&buf[idx];
      const float4* g = src + tileBase + idx;
      asm volatile("global_load_async_to_lds_b128 %0, %1, off"
                   :
                   : "v"(ldsAddr), "v"(g)
                   : "memory");
    }
    // Drain ASYNCcnt: all 16 per-wave loads landed in this wave's LDS slots.
    asm volatile("s_wait_asynccnt 0" ::: "memory");

#pragma unroll
    for (int k = 0; k < KBATCH; ++k) {
      const unsigned idx = k * THREADS + t;
      const unsigned ldsAddr = (unsigned)(uintptr_t)&buf[idx];
      float4* g = dst + tileBase + idx;
      asm volatile("global_store_async_from_lds_b128 %0, %1, off"
                   :
                   : "v"(g), "v"(ldsAddr)
                   : "memory");
    }
    // Ensure LDS->memory stores retire before wave teardown.
    asm volatile("s_wait_asynccnt 0" ::: "memory");
  } else {
    // ---- tail tile: guarded plain copy (no async ops under divergence) ----
    for (int k = 0; k < KBATCH; ++k) {
      const long long idx = tileBase + k * THREADS + t;
      if (idx < n4) dst[idx] = src[idx];
    }
  }
}

// Handles the final (n % 4) scalar floats, if any.
__global__ void copy_tail_kernel(const float* __restrict__ src,
                                 float* __restrict__ dst,
                                 long long start, long long n) {
  long long i = start + (long long)blockIdx.x * blockDim.x + threadIdx.x;
  if (i < n) dst[i] = src[i];
}

extern "C" void kernel_launch(void* const* d_in, const int* in_sizes, int n_in,
                              void* d_out, int out_size, void* d_ws, size_t ws_size,
                              hipStream_t stream) {
  (void)n_in; (void)d_ws; (void)ws_size; (void)in_sizes;

  const float* x = (const float*)d_in[0];
  float* out = (float*)d_out;

  const long long n = (long long)out_size;  // total float elements
  const long long n4 = n / 4;               // float4 elements
  const long long rem = n - n4 * 4;

  if (n4 > 0) {
    const long long tiles = (n4 + TILE4 - 1) / TILE4;  // 4096 for 64 x 2^20
    copy_async_lds_kernel<<<dim3((unsigned)tiles), dim3(THREADS), 0, stream>>>(
        (const float4*)x, (float4*)out, n4);
  }
  if (rem > 0) {
    copy_tail_kernel<<<1, 64, 0, stream>>>(x, out, n4 * 4, n);
  }
}